// Net_50543175139451
// MI455X (gfx1250) — compile-verified
//
#include <hip/hip_runtime.h>
#include <math.h>

// ---------------------------------------------------------------------------
// Capsule routing: per-row softmax(b) dotted with u_hat, then global squash.
//   b: [CAPS=4096][J=8192] f32, u_hat: [J] f32  ->  v: [CAPS] f32
//
// Bandwidth-bound (134 MB / 23.3 TB/s ~ 5.8 us). Phase A computes row maxes
// (streams b from HBM into L2; b fits in 192MB L2), Phase B re-reads the same
// rows (L2 hits), computes exp in f32, and fuses BOTH softmax reductions
// (sum exp*u and sum exp) into one V_WMMA_F32_16X16X4_F32 per 16x4 tile:
//   A = exp(b - rowmax) tile, B col0 = u_hat chunk, B col1 = ones.
//   D[:,0] -> numerators, D[:,1] -> denominators. Full f32 precision.
//
// All loops use compile-time-constant trip counts so control flow is uniform
// and EXEC stays all-ones across every WMMA (ISA 7.12 requirement). The B
// fragment is built with per-lane 0/1 multipliers (fmaf) so the u_hat value
// is consumed by every lane -> the compiler cannot sink its load under a
// divergent exec mask.
// ---------------------------------------------------------------------------

typedef __attribute__((ext_vector_type(2))) float v2f;
typedef __attribute__((ext_vector_type(8))) float v8f;

#define J_DIM   8192
#define CAPS    4096
#define NWAVES  8                 // waves per workgroup
#define KSLICE  (J_DIM / NWAVES)  // 1024 columns per wave

__global__ __launch_bounds__(256) void caps_softmax_matvec_kernel(
    const float* __restrict__ u_hat,
    const float* __restrict__ b,
    float* __restrict__ s_out)
{
    __shared__ float lds_u[J_DIM];         // staged u_hat (wave-private slices)
    __shared__ float smax[16][NWAVES];
    __shared__ float snum[16][NWAVES];
    __shared__ float sden[16][NWAVES];

    const int tid  = threadIdx.x;
    const int lane = tid & 31;
    const int wave = tid >> 5;
    const int col  = lane & 15;   // N-column of B/D; also row-in-tile for A
    const int half = lane >> 4;   // 0: K={0,1}, 1: K={2,3} (f32 WMMA layout)

    const int    row0      = blockIdx.x * 16;
    const float* tile_base = b + (size_t)row0 * J_DIM;
    const int    k0        = wave * KSLICE;   // this wave's K-slice

    // ---- Stage this wave's u_hat slice into LDS (read back by same wave) --
    {
        const float* src = u_hat + k0 + lane * 4;
        float*       dst = lds_u + k0 + lane * 4;
        #pragma unroll
        for (int it = 0; it < KSLICE / (32 * 4); ++it)        // 8 iters
            *(float4*)(dst + it * 128) = *(const float4*)(src + it * 128);
    }

    // ---------------- Phase A: per-row max over this wave's K-slice --------
    for (int m = 0; m < 16; ++m) {
        const float* r = tile_base + (size_t)m * J_DIM + k0 + lane * 4;
        float mx = -INFINITY;
        #pragma unroll
        for (int it = 0; it < KSLICE / (32 * 4); ++it) {      // 8 iters, coalesced
            float4 v4 = *(const float4*)(r + it * 128);
            mx = fmaxf(mx, fmaxf(fmaxf(v4.x, v4.y), fmaxf(v4.z, v4.w)));
        }
        #pragma unroll
        for (int off = 16; off > 0; off >>= 1)
            mx = fmaxf(mx, __shfl_xor(mx, off, 32));
        if (lane == 0) smax[m][wave] = mx;
    }
    __syncthreads();

    // Full-row max for the row this lane feeds into the A-fragment (row = col).
    float mymax = -INFINITY;
    #pragma unroll
    for (int w = 0; w < NWAVES; ++w) mymax = fmaxf(mymax, smax[col][w]);

    // ---------------- Phase B: exp + WMMA accumulate over the K-slice ------
    // A fragment (16x4 f32): lane holds b[row0+col][kb + 2*half + {0,1}]
    // B fragment (4x16 f32): VGPR0 -> K=half, VGPR1 -> K=2+half, N = col
    //   B[k][0] = u_hat[kb+k], B[k][1] = 1.0, else 0.
    const float sel_u   = (col == 0) ? 1.0f : 0.0f;
    const float sel_one = (col == 1) ? 1.0f : 0.0f;
    const float* myrow  = tile_base + (size_t)col * J_DIM + k0 + 2 * half;
    const float* ubase  = lds_u + k0 + half;   // 'half' folded into LDS address

    v8f acc = {};
    #pragma unroll 4
    for (int it = 0; it < KSLICE / 4; ++it) {                 // 256 iters, uniform
        v2f bv = *(const v2f*)(myrow + it * 4);
        v2f a;
        a.x = __expf(bv.x - mymax);
        a.y = __expf(bv.y - mymax);

        // Broadcast LDS reads (lanes 0-15 one address, 16-31 the next): no
        // bank conflicts, consumed by ALL lanes via fmaf -> stays unpredicated.
        float u_lo = ubase[it * 4];            // u_hat[kb + half]
        float u_hi = ubase[it * 4 + 2];        // u_hat[kb + 2 + half]
        v2f bf;
        bf.x = fmaf(sel_u, u_lo, sel_one);
        bf.y = fmaf(sel_u, u_hi, sel_one);

        // D = A x B + C  (f32 accumulate; EXEC all-ones)
        acc = __builtin_amdgcn_wmma_f32_16x16x4_f32(
            /*neg_a=*/false, a, /*neg_b=*/false, bf,
            /*c_mod=*/(short)0, acc, /*reuse_a=*/false, /*reuse_b=*/false);
    }

    // ---------------- Extract D columns 0 (num) and 1 (den) ----------------
    // D[m][n]: VGPR v -> M = v + 8*half (N = col). num in lanes 0/16, den 1/17.
    if (col == 0 || col == 1) {
        float* dst = (col == 0) ? &snum[0][0] : &sden[0][0];
        #pragma unroll
        for (int v = 0; v < 8; ++v) {
            int m = v + 8 * half;
            dst[m * NWAVES + wave] = acc[v];
        }
    }
    __syncthreads();

    // Combine wave partials; s = num / den  (softmax-weighted dot, exact)
    if (tid < 16) {
        float num = 0.0f, den = 0.0f;
        #pragma unroll
        for (int w = 0; w < NWAVES; ++w) { num += snum[tid][w]; den += sden[tid][w]; }
        s_out[row0 + tid] = num / den;
    }
}

// ---------------- Kernel 2: global squash over s[4096] ---------------------
__global__ __launch_bounds__(256) void caps_squash_kernel(
    const float* __restrict__ s, float* __restrict__ v_out)
{
    __shared__ float red[256];
    float acc = 0.0f;
    for (int i = threadIdx.x; i < CAPS; i += 256) {
        float x = s[i];
        acc += x * x;
    }
    red[threadIdx.x] = acc;
    __syncthreads();
    #pragma unroll
    for (int o = 128; o > 0; o >>= 1) {
        if (threadIdx.x < (unsigned)o) red[threadIdx.x] += red[threadIdx.x + o];
        __syncthreads();
    }
    const float msq   = red[0];
    const float scale = msq / ((1.0f + msq) * sqrtf(msq));
    for (int i = threadIdx.x; i < CAPS; i += 256) v_out[i] = scale * s[i];
}

extern "C" void kernel_launch(void* const* d_in, const int* in_sizes, int n_in,
                              void* d_out, int out_size, void* d_ws, size_t ws_size,
                              hipStream_t stream) {
    const float* u_hat = (const float*)d_in[0];   // [J_DIM]
    const float* b     = (const float*)d_in[1];   // [CAPS][J_DIM]
    float* s = (float*)d_ws;                      // [CAPS] scratch
    float* v = (float*)d_out;                     // [CAPS] output

    caps_softmax_matvec_kernel<<<CAPS / 16, 256, 0, stream>>>(u_hat, b, s);
    caps_squash_kernel<<<1, 256, 0, stream>>>(s, v);
}